// RPPsAtt_74698071212199
// MI455X (gfx1250) — compile-verified
//
#include <hip/hip_runtime.h>
#include <hip/hip_bf16.h>

// Problem constants (reference: N=512, Q=1024, D=256, fp32 everywhere).
#define NN 512
#define QQ 1024
#define DD 256

typedef __attribute__((ext_vector_type(2))) float v2f;
typedef __attribute__((ext_vector_type(8))) float v8f;

// ---------------------------------------------------------------------------
// Kernel 1: q = query @ W_in^T  via V_WMMA_F32_16X16X4_F32.
// One wave (32 threads) computes one 16x16 C tile, K stepped by 4.
// Writes q into combined[b*2D + 256 + n] (second half of the concat operand).
// A layout (ISA 7.12.2, 32-bit A 16x4): lanes 0-15 -> M rows, VGPR0=K, VGPR1=K+1;
// lanes 16-31 -> same M rows, K+2/K+3. B mirrored; C: VGPR i -> M=i (+8 for hi half).
// ---------------------------------------------------------------------------
__global__ __launch_bounds__(32) void qproj_wmma(const float* __restrict__ query,
                                                 const float* __restrict__ W_in,
                                                 float* __restrict__ combined) {
    const int mtile = blockIdx.x;          // batch tile (N/16)
    const int ntile = blockIdx.y;          // dout tile (D/16)
    const int lane  = threadIdx.x;
    const int hi    = lane >> 4;           // 0: K,K+1   1: K+2,K+3
    const int l15   = lane & 15;           // M row for A, N col for B

    const float* arow = query + (size_t)(mtile * 16 + l15) * DD;   // A row (M)
    const float* brow = W_in  + (size_t)(ntile * 16 + l15) * DD;   // B col n = W_in[n, k]

    v8f c = {};
    for (int k0 = 0; k0 < DD; k0 += 4) {
        const int ka = k0 + hi * 2;
        v2f a, b;
        a.x = arow[ka];     a.y = arow[ka + 1];
        b.x = brow[ka];     b.y = brow[ka + 1];
        c = __builtin_amdgcn_wmma_f32_16x16x4_f32(false, a, false, b,
                                                  (short)0, c, false, false);
    }
    // Store C tile: VGPR i -> row M = i + 8*hi, col = l15.
    float* outbase = combined + (size_t)(mtile * 16) * (2 * DD) + DD + ntile * 16 + l15;
#pragma unroll
    for (int i = 0; i < 8; ++i) {
        outbase[(size_t)(i + hi * 8) * (2 * DD)] = c[i];
    }
}

// ---------------------------------------------------------------------------
// Kernel 2: fused attention, single pass over context (flash-style online
// softmax). One 256-thread block (8 waves) per node b. Each lane owns 8
// contiguous d-columns (two coalesced float4 loads per context row).
//   A1[d] = sum_q p_q * c[q,d]
//   A2[d] = sum_q p_q * bt_q * |c[q,d]|,  bt_q = exp(-ab*(Q-1-q))
//   mix2[d] = tanh((A1 + (|ae|-|ad|)*A2)/L)        -> combined[b*2D + d]
//   attn[b,q] = exp(s_q - M)/L                     -> d_out attention region
// ---------------------------------------------------------------------------
__global__ __launch_bounds__(256) void attention_fused(
        const float* __restrict__ context,   // [N,Q,D]
        const float* __restrict__ ae,
        const float* __restrict__ ad,
        const float* __restrict__ ab,
        float* __restrict__ combined,        // [N,2D]; q preloaded at +DD
        float* __restrict__ attn_out) {      // [N,Q]
    const int b    = blockIdx.x;
    const int tid  = threadIdx.x;
    const int wave = tid >> 5;               // 0..7
    const int lane = tid & 31;
    const int dbase = lane * 8;              // this lane's 8 d-columns

    __shared__ float q_s[DD];
    __shared__ float scores_s[QQ];
    __shared__ float wm[8], wl[8];
    __shared__ float wacc1[8 * DD];
    __shared__ float wacc2[8 * DD];

    q_s[tid] = combined[(size_t)b * (2 * DD) + DD + tid];
    __syncthreads();

    float qreg[8];
#pragma unroll
    for (int j = 0; j < 8; ++j) qreg[j] = q_s[dbase + j];

    const float A_b = ab[b];
    const float* ctx_b = context + (size_t)b * QQ * DD;

    float m = -__builtin_inff();
    float l = 0.0f;
    float a1[8], a2[8];
#pragma unroll
    for (int j = 0; j < 8; ++j) { a1[j] = 0.0f; a2[j] = 0.0f; }

    const int q0 = wave * (QQ / 8);
    for (int qi = q0; qi < q0 + (QQ / 8); ++qi) {
        const float* row = ctx_b + (size_t)qi * DD + dbase;
        __builtin_prefetch(row + 8 * DD, 0, 1);   // global_prefetch_b8, 8 rows ahead
        float4 c0 = *(const float4*)(row);
        float4 c1 = *(const float4*)(row + 4);
        float cr[8] = {c0.x, c0.y, c0.z, c0.w, c1.x, c1.y, c1.z, c1.w};

        float dot = 0.0f;
#pragma unroll
        for (int j = 0; j < 8; ++j) dot += qreg[j] * cr[j];
        // wave32 all-reduce
#pragma unroll
        for (int off = 16; off >= 1; off >>= 1) dot += __shfl_xor(dot, off, 32);
        const float s = dot;
        if (lane == 0) scores_s[qi] = s;

        const float nm    = fmaxf(m, s);
        const float scale = expf(m - nm);          // m=-inf first iter -> 0
        const float p     = expf(s - nm);
        const float bt    = expf(-A_b * (float)(QQ - 1 - qi));
        const float pb    = p * bt;
        l = l * scale + p;
#pragma unroll
        for (int j = 0; j < 8; ++j) {
            a1[j] = a1[j] * scale + p  * cr[j];
            a2[j] = a2[j] * scale + pb * fabsf(cr[j]);
        }
        m = nm;
    }

    if (lane == 0) { wm[wave] = m; wl[wave] = l; }
#pragma unroll
    for (int j = 0; j < 8; ++j) {
        wacc1[wave * DD + dbase + j] = a1[j];
        wacc2[wave * DD + dbase + j] = a2[j];
    }
    __syncthreads();

    // Merge the 8 per-wave softmax states (redundantly in every thread: cheap).
    float M = wm[0];
#pragma unroll
    for (int w = 1; w < 8; ++w) M = fmaxf(M, wm[w]);
    float L = 0.0f;
    float esc[8];
#pragma unroll
    for (int w = 0; w < 8; ++w) { esc[w] = expf(wm[w] - M); L += wl[w] * esc[w]; }
    const float invL = 1.0f / L;

    // Thread tid owns d = tid.
    float A1 = 0.0f, A2 = 0.0f;
#pragma unroll
    for (int w = 0; w < 8; ++w) {
        A1 += esc[w] * wacc1[w * DD + tid];
        A2 += esc[w] * wacc2[w * DD + tid];
    }
    const float fa = fabsf(ae[b]);
    const float fd = fabsf(ad[b]);
    combined[(size_t)b * (2 * DD) + tid] = tanhf((A1 + (fa - fd) * A2) * invL);

    // Normalized attention weights out.
    for (int qi = tid; qi < QQ; qi += 256) {
        attn_out[(size_t)b * QQ + qi] = expf(scores_s[qi] - M) * invL;
    }
}

// ---------------------------------------------------------------------------
// Kernel 3: output = tanh(combined @ W_out^T), GEMM 512x256x512 via
// V_WMMA_F32_16X16X4_F32, same tiling as kernel 1.
// ---------------------------------------------------------------------------
__global__ __launch_bounds__(32) void outproj_wmma(const float* __restrict__ combined,
                                                   const float* __restrict__ W_out,
                                                   float* __restrict__ out) {
    const int mtile = blockIdx.x;          // batch tile
    const int ntile = blockIdx.y;          // dout tile
    const int lane  = threadIdx.x;
    const int hi    = lane >> 4;
    const int l15   = lane & 15;

    const float* arow = combined + (size_t)(mtile * 16 + l15) * (2 * DD);
    const float* brow = W_out    + (size_t)(ntile * 16 + l15) * (2 * DD);  // W_out[n, k]

    v8f c = {};
    for (int k0 = 0; k0 < 2 * DD; k0 += 4) {
        const int ka = k0 + hi * 2;
        v2f a, b;
        a.x = arow[ka];     a.y = arow[ka + 1];
        b.x = brow[ka];     b.y = brow[ka + 1];
        c = __builtin_amdgcn_wmma_f32_16x16x4_f32(false, a, false, b,
                                                  (short)0, c, false, false);
    }
    float* outbase = out + (size_t)(mtile * 16) * DD + ntile * 16 + l15;
#pragma unroll
    for (int i = 0; i < 8; ++i) {
        outbase[(size_t)(i + hi * 8) * DD] = tanhf(c[i]);
    }
}

// ---------------------------------------------------------------------------
extern "C" void kernel_launch(void* const* d_in, const int* in_sizes, int n_in,
                              void* d_out, int out_size, void* d_ws, size_t ws_size,
                              hipStream_t stream) {
    const float* query   = (const float*)d_in[0];   // [N,1,D]
    const float* context = (const float*)d_in[1];   // [N,Q,D]
    const float* W_in    = (const float*)d_in[2];   // [D,D]
    const float* W_out   = (const float*)d_in[3];   // [D,2D]
    const float* ae      = (const float*)d_in[4];   // [N,1,1]
    const float* ad      = (const float*)d_in[5];
    const float* ab      = (const float*)d_in[6];
    // d_in[7] = index, unused in forward math.

    float* out  = (float*)d_out;                    // [N,D] output first
    float* attn = out + (size_t)NN * DD;            // then [N,Q] attention

    // Workspace: combined[N, 2D] (mix2 | q), 1 MB.
    float* combined = (float*)d_ws;

    dim3 gproj(NN / 16, DD / 16);   // (32,16) tiles, one wave per tile
    qproj_wmma<<<gproj, 32, 0, stream>>>(query, W_in, combined);
    attention_fused<<<NN, 256, 0, stream>>>(context, ae, ad, ab, combined, attn);
    outproj_wmma<<<gproj, 32, 0, stream>>>(combined, W_out, out);
}